// BiMambaEncoder_11879879544348
// MI455X (gfx1250) — compile-verified
//
#include <hip/hip_runtime.h>
#include <math.h>

// ---------------------------------------------------------------------------
// BiMamba encoder for MI455X (gfx1250).
// f32 WMMA (V_WMMA_F32_16X16X4_F32) for every GEMM: keeps reference f32
// numerics while engaging the CDNA5 matrix pipe.
//  - gemm_f32_wmma_t64: 64x64 block tile, 4 waves, B staged in LDS in
//    K-pair-interleaved layout so each WMMA B operand is one aligned
//    ds_load_b64 into an even VGPR pair (no repack movs); 16 v_wmma per
//    K-chunk; global_prefetch of the next B chunk overlaps compute.
//  - gemm_f32_wmma:     fallback wave-per-16x16-tile for N % 64 != 0 (N=48).
// Scan is the serial part: parallelized over B*ED rows, N=16 states in
// registers, per-step B/C vectors staged in LDS for the whole block.
// ---------------------------------------------------------------------------

typedef float v2f __attribute__((ext_vector_type(2)));
typedef float v8f __attribute__((ext_vector_type(8)));

#define EPSV 1e-5f
#define KC 16
// LDS pair-row stride in floats: 128 used + 32 pad so consecutive K-pairs map
// to opposite 32-bank halves (conflict-free half-wave b64 reads).
#define BSTRIDE 160

__device__ __forceinline__ float siluf(float x) { return x / (1.0f + expf(-x)); }
__device__ __forceinline__ float softplusf(float x) {
    return (x > 20.0f) ? x : log1pf(expf(x));
}

// ---------------------------------------------------------------------------
// Tiled GEMM: C[M,N] = A[M,K](lda) @ B[K,N](ldb) (+bias) (+=C)
// Requires M % 64 == 0, N % 64 == 0, K % 16 == 0.
// Block = 128 threads (4 waves). Block computes a 64x64 tile; wave w owns the
// 16x64 strip [w*16, w*16+16). B is staged per 16-K chunk in LDS, interleaved
// as Bs[pair][col][2] (pair = k/2), and reused by all 4 waves and all 4
// N-subtiles -> 16 v_wmma per chunk iteration.
// ---------------------------------------------------------------------------
__global__ __launch_bounds__(128) void gemm_f32_wmma_t64(
    const float* __restrict__ A, int lda,
    const float* __restrict__ B, int ldb,
    const float* __restrict__ bias,
    float* __restrict__ C, int ldc,
    int M, int N, int K, int accumulate)
{
    __shared__ float Bs[8 * BSTRIDE];      // 8 K-pairs x (64 cols x 2) + pad

    const int tiles_n = N >> 6;
    const int bm = blockIdx.x / tiles_n;
    const int bn = blockIdx.x % tiles_n;
    const int tidx = threadIdx.x;
    const int wave = tidx >> 5;
    const int lane = tidx & 31;
    const int half = lane >> 4;            // 0: K pair (k,k+1)  1: (k+2,k+3)
    const int lm   = lane & 15;

    const int arow = bm * 64 + wave * 16 + lm;
    const float* Arow = A + (size_t)arow * lda + 2 * half;
    const int n0 = bn * 64;

    // cooperative B fill: thread t owns K-pair (t>>4) and 4 columns (t&15)*4
    const int bp   = tidx >> 4;            // 0..7
    const int bcol = (tidx & 15) * 4;      // 0,4,...,60
    float* bdst = &Bs[bp * BSTRIDE + bcol * 2];

    v8f acc[4] = {v8f{}, v8f{}, v8f{}, v8f{}};

    for (int kc = 0; kc < K; kc += KC) {
        __syncthreads();
        {
            const float4 v0 = *(const float4*)(B + (size_t)(kc + 2 * bp)     * ldb + n0 + bcol);
            const float4 v1 = *(const float4*)(B + (size_t)(kc + 2 * bp + 1) * ldb + n0 + bcol);
            ((v2f*)bdst)[0] = v2f{v0.x, v1.x};
            ((v2f*)bdst)[1] = v2f{v0.y, v1.y};
            ((v2f*)bdst)[2] = v2f{v0.z, v1.z};
            ((v2f*)bdst)[3] = v2f{v0.w, v1.w};
        }
        __syncthreads();

        if (kc + KC < K)   // pull next B chunk toward the caches while we compute
            __builtin_prefetch(B + (size_t)(kc + KC + 2 * bp) * ldb + n0 + bcol, 0, 1);

#pragma unroll
        for (int kk = 0; kk < KC; kk += 4) {
            v2f a;
            a.x = Arow[kc + kk];
            a.y = Arow[kc + kk + 1];
            const float* brow = &Bs[(size_t)((kk >> 1) + half) * BSTRIDE];
#pragma unroll
            for (int sub = 0; sub < 4; ++sub) {
                v2f b = *(const v2f*)&brow[(sub * 16 + lm) * 2];
                acc[sub] = __builtin_amdgcn_wmma_f32_16x16x4_f32(
                    false, a, false, b, (short)0, acc[sub], false, false);
            }
        }
    }

#pragma unroll
    for (int sub = 0; sub < 4; ++sub) {
        const int col = n0 + sub * 16 + lm;
        const float bv = bias ? bias[col] : 0.0f;
#pragma unroll
        for (int r = 0; r < 8; ++r) {
            const int orow = bm * 64 + wave * 16 + r + half * 8;
            const size_t idx = (size_t)orow * ldc + col;
            float v = acc[sub][r] + bv;
            if (accumulate) v += C[idx];
            C[idx] = v;
        }
    }
}

// ---------------------------------------------------------------------------
// Fallback GEMM: one wave per 16x16 tile (used when N % 64 != 0, e.g. N=48).
// ---------------------------------------------------------------------------
__global__ __launch_bounds__(128) void gemm_f32_wmma(
    const float* __restrict__ A, int lda,
    const float* __restrict__ B, int ldb,
    const float* __restrict__ bias,
    float* __restrict__ C, int ldc,
    int M, int N, int K, int accumulate)
{
    const int lane = threadIdx.x & 31;
    const int wave = threadIdx.x >> 5;
    const int tiles_n = N >> 4;
    const int tiles_m = M >> 4;
    const int tile = blockIdx.x * 4 + wave;
    if (tile >= tiles_m * tiles_n) return;   // whole wave exits together
    const int tm = tile / tiles_n;
    const int tn = tile % tiles_n;
    const int half = lane >> 4;
    const int lm = lane & 15;

    const float* Arow = A + (size_t)(tm * 16 + lm) * lda + 2 * half;
    const float* Bcol = B + (size_t)(2 * half) * ldb + tn * 16 + lm;

    v8f acc = {};
    for (int k0 = 0; k0 < K; k0 += 4) {
        v2f a;
        a.x = Arow[k0];
        a.y = Arow[k0 + 1];
        v2f b;
        b.x = Bcol[(size_t)k0 * ldb];
        b.y = Bcol[(size_t)(k0 + 1) * ldb];
        acc = __builtin_amdgcn_wmma_f32_16x16x4_f32(
            false, a, false, b, (short)0, acc, false, false);
    }

    const int col = tn * 16 + lm;
    const float bv = bias ? bias[col] : 0.0f;
#pragma unroll
    for (int r = 0; r < 8; ++r) {
        const int row = tm * 16 + r + half * 8;
        const size_t idx = (size_t)row * ldc + col;
        float v = acc[r] + bv;
        if (accumulate) v += C[idx];
        C[idx] = v;
    }
}

// feat (B,C,H,W) -> featT (B*L, C) row-major, L = H*W
__global__ void transpose_feat(const float* __restrict__ feat,
                               float* __restrict__ out,
                               int Bsz, int Cch, int Ltok)
{
    int i = blockIdx.x * blockDim.x + threadIdx.x;
    int total = Bsz * Ltok * Cch;
    if (i >= total) return;
    int c = i % Cch;
    int t = i / Cch;
    int l = t % Ltok;
    int b = t / Ltok;
    out[i] = feat[((size_t)b * Cch + c) * Ltok + l];
}

__global__ void add_pos(float* __restrict__ x, const float* __restrict__ pos,
                        int Bsz, int Ltok, int dim)
{
    int i = blockIdx.x * blockDim.x + threadIdx.x;
    int total = Bsz * Ltok * dim;
    if (i >= total) return;
    int d = i % dim;
    int l = (i / dim) % Ltok;
    x[i] += pos[(size_t)l * dim + d];
}

__global__ void copy_buf(const float* __restrict__ a, float* __restrict__ b, int n)
{
    int i = blockIdx.x * blockDim.x + threadIdx.x;
    if (i < n) b[i] = a[i];
}

// dst[b,l,:] (=|+=) src[b,L-1-l,:]
__global__ void flip_op(const float* __restrict__ src, float* __restrict__ dst,
                        int Bsz, int Ltok, int dim, int add)
{
    int i = blockIdx.x * blockDim.x + threadIdx.x;
    int total = Bsz * Ltok * dim;
    if (i >= total) return;
    int d = i % dim;
    int t = i / dim;
    int l = t % Ltok;
    int b = t / Ltok;
    float v = src[((size_t)(b * Ltok + (Ltok - 1 - l))) * dim + d];
    if (add) dst[i] += v;
    else     dst[i] = v;
}

// RMSNorm over dim; one block (256 == dim) per token
__global__ __launch_bounds__(256) void rmsnorm_k(
    const float* __restrict__ x, const float* __restrict__ w,
    float* __restrict__ o, int dim)
{
    __shared__ float red[256];
    const int t = threadIdx.x;
    const size_t base = (size_t)blockIdx.x * dim;
    float v = x[base + t];
    red[t] = v * v;
    __syncthreads();
    for (int s = 128; s > 0; s >>= 1) {
        if (t < s) red[t] += red[t + s];
        __syncthreads();
    }
    float r = rsqrtf(red[0] / dim + EPSV);
    o[base + t] = v * r * w[t];
}

// Depthwise causal conv (K=4, cross-correlation, left pad 3) + SiLU.
// Input = xz[:, :ED] (row stride 2*ED), output xc (row stride ED).
__global__ void conv_silu(const float* __restrict__ xz,
                          const float* __restrict__ cw,
                          const float* __restrict__ cb,
                          float* __restrict__ xc,
                          int Bsz, int Ltok, int ED)
{
    int i = blockIdx.x * blockDim.x + threadIdx.x;
    int total = Bsz * Ltok * ED;
    if (i >= total) return;
    int e = i % ED;
    int t = i / ED;
    int l = t % Ltok;
    int b = t / Ltok;
    float s = cb[e];
#pragma unroll
    for (int k = 0; k < 4; ++k) {
        int ls = l + k - 3;
        if (ls >= 0)
            s += cw[e * 4 + k] * xz[((size_t)(b * Ltok + ls)) * (2 * ED) + e];
    }
    xc[i] = siluf(s);
}

// Selective scan. delta_raw holds (dr@wdt + bdt) pre-softplus.
// dbl row layout: [0:DR)=dr, [DR:DR+N)=Bvec, [DR+N:DR+2N)=Cvec, width 48.
// Block: 256 threads = 256 channels of one batch; grid = B * (ED/256).
__global__ __launch_bounds__(256) void scan_k(
    const float* __restrict__ delta_raw, const float* __restrict__ xc,
    const float* __restrict__ dbl, const float* __restrict__ Alog,
    const float* __restrict__ Dp, float* __restrict__ y,
    int Bsz, int Ltok, int ED)
{
    const int N = 16, DR = 16, DBLW = 48;
    const int chunks = ED / 256;
    const int b = blockIdx.x / chunks;
    const int e = (blockIdx.x % chunks) * 256 + threadIdx.x;

    float h[16];
    float Ae[16];
#pragma unroll
    for (int n = 0; n < 16; ++n) {
        h[n] = 0.0f;
        Ae[n] = -expf(Alog[(size_t)e * N + n]);
    }
    const float De = Dp[e];

    __shared__ float sB[16];
    __shared__ float sC[16];

    for (int l = 0; l < Ltok; ++l) {
        const size_t m = (size_t)b * Ltok + l;
        __syncthreads();
        if (threadIdx.x < 32) {
            float v = dbl[m * DBLW + DR + threadIdx.x];
            if (threadIdx.x < 16) sB[threadIdx.x] = v;
            else                  sC[threadIdx.x - 16] = v;
        }
        __syncthreads();
        const float dlt = softplusf(delta_raw[m * ED + e]);
        const float xv  = xc[m * ED + e];
        float acc = 0.0f;
#pragma unroll
        for (int n = 0; n < 16; ++n) {
            float dA = expf(dlt * Ae[n]);
            h[n] = dA * h[n] + dlt * sB[n] * xv;
            acc += h[n] * sC[n];
        }
        y[m * ED + e] = acc + De * xv;
    }
}

// y *= silu(z), z = xz[:, ED + e]
__global__ void gate_silu(float* __restrict__ y, const float* __restrict__ xz,
                          int ED, int total)
{
    int i = blockIdx.x * blockDim.x + threadIdx.x;
    if (i >= total) return;
    int e = i % ED;
    int m = i / ED;
    float z = xz[(size_t)m * 2 * ED + ED + e];
    y[i] *= siluf(z);
}

// LayerNorm over dim; one block (256 == dim) per token
__global__ __launch_bounds__(256) void layernorm_k(
    const float* __restrict__ x, const float* __restrict__ w,
    const float* __restrict__ bb, float* __restrict__ o, int dim)
{
    __shared__ float r1[256];
    __shared__ float r2[256];
    const int t = threadIdx.x;
    const size_t base = (size_t)blockIdx.x * dim;
    float v = x[base + t];
    r1[t] = v;
    __syncthreads();
    for (int s = 128; s > 0; s >>= 1) {
        if (t < s) r1[t] += r1[t + s];
        __syncthreads();
    }
    float m = r1[0] / dim;
    float d = v - m;
    r2[t] = d * d;
    __syncthreads();
    for (int s = 128; s > 0; s >>= 1) {
        if (t < s) r2[t] += r2[t + s];
        __syncthreads();
    }
    float var = r2[0] / dim;
    o[base + t] = d * rsqrtf(var + EPSV) * w[t] + bb[t];
}

// out[b,d] = mean_l ln[b,l,d]
__global__ void mean_tokens(const float* __restrict__ ln, float* __restrict__ out,
                            int Bsz, int Ltok, int dim)
{
    int i = blockIdx.x * blockDim.x + threadIdx.x;
    if (i >= Bsz * dim) return;
    int d = i % dim;
    int b = i / dim;
    float s = 0.0f;
    for (int l = 0; l < Ltok; ++l)
        s += ln[((size_t)(b * Ltok + l)) * dim + d];
    out[i] = s / (float)Ltok;
}

// ---------------------------------------------------------------------------
extern "C" void kernel_launch(void* const* d_in, const int* in_sizes, int n_in,
                              void* d_out, int out_size, void* d_ws, size_t ws_size,
                              hipStream_t stream)
{
    (void)in_sizes; (void)n_in; (void)out_size; (void)ws_size;

    const int B = 4, C = 512, L = 576, DIM = 256;
    const int ED = 512, N = 16, DR = 16, NL = 2;
    const int M = B * L;                 // 2304 rows (36 * 64)

    const float* feat   = (const float*)d_in[0];
    const float* proj_w = (const float*)d_in[1];
    const float* proj_b = (const float*)d_in[2];
    const float* posemb = (const float*)d_in[3];
    const float* ln_w   = (const float*)d_in[4];
    const float* ln_b   = (const float*)d_in[5];
    // per-direction weight packs: [win, convw, convb, wx, wdt, bdt, Alog, D, wout, rms]
    const float* const* mf = (const float* const*)&d_in[6];
    const float* const* mb = (const float* const*)&d_in[16];
    float* out = (float*)d_out;

    // workspace partition (floats)
    float* ws    = (float*)d_ws;
    float* featT = ws;                 ws += (size_t)M * C;        // 1,179,648
    float* x     = ws;                 ws += (size_t)M * DIM;      //   589,824
    float* xdir  = ws;                 ws += (size_t)M * DIM;
    float* xn    = ws;                 ws += (size_t)M * DIM;      // also LN buffer
    float* xz    = ws;                 ws += (size_t)M * 2 * ED;   // 2,359,296
    float* xc    = ws;                 ws += (size_t)M * ED;
    float* dbl   = ws;                 ws += (size_t)M * 48;
    float* delta = ws;                 ws += (size_t)M * ED;
    float* y     = ws;                 ws += (size_t)M * ED;
    float* ysum  = ws;                 ws += (size_t)M * DIM;

    auto cdiv = [](int a, int b) { return (a + b - 1) / b; };
    auto gemm = [&](const float* A, int lda, const float* Bm, int ldb,
                    const float* bias, float* Cm, int ldc,
                    int Mm, int Nn, int Kk, int acc) {
        if ((Nn & 63) == 0 && (Mm & 63) == 0 && (Kk & 15) == 0) {
            int tiles = (Mm / 64) * (Nn / 64);
            gemm_f32_wmma_t64<<<tiles, 128, 0, stream>>>(
                A, lda, Bm, ldb, bias, Cm, ldc, Mm, Nn, Kk, acc);
        } else {
            int tiles = (Mm / 16) * (Nn / 16);
            gemm_f32_wmma<<<cdiv(tiles, 4), 128, 0, stream>>>(
                A, lda, Bm, ldb, bias, Cm, ldc, Mm, Nn, Kk, acc);
        }
    };

    // ---- stem: proj + pos ----
    transpose_feat<<<cdiv(M * C, 256), 256, 0, stream>>>(feat, featT, B, C, L);
    gemm(featT, C, proj_w, DIM, proj_b, x, DIM, M, DIM, C, 0);
    add_pos<<<cdiv(M * DIM, 256), 256, 0, stream>>>(x, posemb, B, L, DIM);

    // ---- one mamba layer operating in-place on xd (residual) ----
    auto run_layer = [&](const float* const* p, int li, float* xd) {
        const float* win   = p[0] + (size_t)li * DIM * 2 * ED;
        const float* convw = p[1] + (size_t)li * ED * 4;
        const float* convb = p[2] + (size_t)li * ED;
        const float* wx    = p[3] + (size_t)li * ED * 48;
        const float* wdt   = p[4] + (size_t)li * DR * ED;
        const float* bdt   = p[5] + (size_t)li * ED;
        const float* Alog  = p[6] + (size_t)li * ED * N;
        const float* Dp    = p[7] + (size_t)li * ED;
        const float* wout  = p[8] + (size_t)li * ED * DIM;
        const float* rmsw  = p[9] + (size_t)li * DIM;

        rmsnorm_k<<<M, 256, 0, stream>>>(xd, rmsw, xn, DIM);
        gemm(xn, DIM, win, 2 * ED, nullptr, xz, 2 * ED, M, 2 * ED, DIM, 0);
        conv_silu<<<cdiv(M * ED, 256), 256, 0, stream>>>(xz, convw, convb, xc, B, L, ED);
        gemm(xc, ED, wx, 48, nullptr, dbl, 48, M, 48, ED, 0);
        gemm(dbl, 48, wdt, ED, bdt, delta, ED, M, ED, DR, 0);   // dr = dbl[:, :DR]
        scan_k<<<B * (ED / 256), 256, 0, stream>>>(delta, xc, dbl, Alog, Dp, y, B, L, ED);
        gate_silu<<<cdiv(M * ED, 256), 256, 0, stream>>>(y, xz, ED, M * ED);
        gemm(y, ED, wout, DIM, nullptr, xd, DIM, M, DIM, ED, 1); // residual add
    };

    // ---- forward direction ----
    copy_buf<<<cdiv(M * DIM, 256), 256, 0, stream>>>(x, xdir, M * DIM);
    for (int li = 0; li < NL; ++li) run_layer(mf, li, xdir);
    copy_buf<<<cdiv(M * DIM, 256), 256, 0, stream>>>(xdir, ysum, M * DIM);

    // ---- backward direction (flip in, flip out) ----
    flip_op<<<cdiv(M * DIM, 256), 256, 0, stream>>>(x, xdir, B, L, DIM, 0);
    for (int li = 0; li < NL; ++li) run_layer(mb, li, xdir);
    flip_op<<<cdiv(M * DIM, 256), 256, 0, stream>>>(xdir, ysum, B, L, DIM, 1);

    // ---- head: LN + mean over tokens ----
    layernorm_k<<<M, 256, 0, stream>>>(ysum, ln_w, ln_b, xn, DIM);
    mean_tokens<<<cdiv(B * DIM, 256), 256, 0, stream>>>(xn, out, B, L, DIM);
}